// LocalizationLoss_10179072492119
// MI455X (gfx1250) — compile-verified
//
#include <hip/hip_runtime.h>
#include <math.h>

// ---------------------------------------------------------------------------
// LocalizationLoss (Hungarian matching + bce/mse/ce loss) for MI455X (gfx1250)
//   - 1 workgroup per batch (64 blocks x 128 threads = 4 wave32)
//   - cost matrix via V_WMMA_F32_16X16X4_F32 (Gram expansion of cdist)
//   - Jonker-Volgenant Hungarian in LDS, f64 potentials (mirrors reference)
//   - deterministic two-stage reduction (no float atomics)
// ---------------------------------------------------------------------------

typedef __attribute__((ext_vector_type(2))) float v2f;
typedef __attribute__((ext_vector_type(8))) float v8f;

#define BN   128   // boxes per batch
#define OC   21    // output channels: 1 presence + 3 box + 17 "logits" (ref uses 4:)
#define TC   6     // target channels: 1 presence + 3 box + 1 class (+pad col)
#define NCLS 17    // reference slices output[...,4:] -> 17 logits (5+16-4)

#define ALPHA 1.0f
#define BETA  5.0f
#define GAMMA 1.0f

// LDS carve (dynamic shared):
//   doubles: u[132], v[132], minv[132], redv[128]
//   floats : cost[128*128], bpS[128*4], btS[128*4], nbp[128], nbt[128]
//   ints   : p[132], way[132], used[132], redj[128], ctrl[4]
static constexpr int    SMEM_DBL   = 132 * 3 + 128;
static constexpr int    SMEM_FLT   = BN * BN + 512 + 512 + 128 + 128;
static constexpr int    SMEM_INT   = 132 * 3 + 128 + 4;
static constexpr size_t SMEM_BYTES = (size_t)SMEM_DBL * 8 + (size_t)SMEM_FLT * 4 + (size_t)SMEM_INT * 4;

__global__ void __launch_bounds__(128)
hungarian_loss_kernel(const float* __restrict__ outp,
                      const float* __restrict__ tgt,
                      double* __restrict__ batch_loss)
{
    extern __shared__ unsigned char smem_raw[];
    double* u    = (double*)smem_raw;   // [132] potentials (rows), index 0..128
    double* v    = u + 132;             // [132] potentials (cols)
    double* minv = v + 132;             // [132]
    double* redv = minv + 132;          // [128] reduction values
    float*  cost = (float*)(redv + 128);   // [128*128]
    float*  bpS  = cost + BN * BN;         // [128*4] output boxes, K padded to 4
    float*  btS  = bpS + 512;              // [128*4] target boxes
    float*  nbp  = btS + 512;              // [128]   |a_i|^2
    float*  nbt  = nbp + 128;              // [128]   |b_j|^2
    int*    pA   = (int*)(nbt + 128);      // [132] column -> matched row (1-based)
    int*    wayA = pA + 132;               // [132]
    int*    usedA= wayA + 132;             // [132]
    int*    redj = usedA + 132;            // [128] reduction indices
    int*    ctrl = redj + 128;             // [0] = j0

    const int b   = blockIdx.x;
    const int tid = threadIdx.x;

    const float* ob_base = outp + (size_t)b * BN * OC;
    const float* tb_base = tgt  + (size_t)b * BN * TC;

    // ---- Phase 1: stage boxes into LDS (K padded 3->4), compute norms ------
    {
        const int i = tid;
        float x0 = ob_base[i * OC + 1];
        float x1 = ob_base[i * OC + 2];
        float x2 = ob_base[i * OC + 3];
        bpS[i * 4 + 0] = x0; bpS[i * 4 + 1] = x1;
        bpS[i * 4 + 2] = x2; bpS[i * 4 + 3] = 0.0f;
        nbp[i] = x0 * x0 + x1 * x1 + x2 * x2;

        float y0 = tb_base[i * TC + 1];
        float y1 = tb_base[i * TC + 2];
        float y2 = tb_base[i * TC + 3];
        btS[i * 4 + 0] = y0; btS[i * 4 + 1] = y1;
        btS[i * 4 + 2] = y2; btS[i * 4 + 3] = 0.0f;
        nbt[i] = y0 * y0 + y1 * y1 + y2 * y2;
    }
    __syncthreads();

    // ---- Phase 2: cost matrix via WMMA f32 16x16x4 --------------------------
    // dot[i][j] = a_i . b_j as 8x8 grid of 16x16 tiles; 16 tiles per wave.
    // A (16x4 f32) layout: lanes 0-15 M=0..15, VGPR0=K0 VGPR1=K1; lanes 16-31 K2/K3.
    // B (4x16 f32) mirrored: lanes 0-15 N=0..15, VGPR0=K0 VGPR1=K1; lanes 16-31 K2/K3.
    {
        const int wave = tid >> 5;
        const int lane = tid & 31;
        const int hi   = lane >> 4;   // 0: K0/K1  1: K2/K3
        const int l16  = lane & 15;
        const int k0   = hi ? 2 : 0;

        for (int t = wave * 16; t < wave * 16 + 16; ++t) {
            const int ti = t >> 3, tj = t & 7;
            const int r0 = ti * 16, c0 = tj * 16;

            v2f a, bv;
            a.x  = bpS[(r0 + l16) * 4 + k0];
            a.y  = bpS[(r0 + l16) * 4 + k0 + 1];
            bv.x = btS[(c0 + l16) * 4 + k0];
            bv.y = btS[(c0 + l16) * 4 + k0 + 1];

            v8f c = {};
            c = __builtin_amdgcn_wmma_f32_16x16x4_f32(
                    false, a, false, bv, (short)0, c, false, false);

            #pragma unroll
            for (int vv = 0; vv < 8; ++vv) {
                const int m = r0 + vv + (hi ? 8 : 0);
                const int n = c0 + l16;
                float d2 = nbp[m] + nbt[n] - 2.0f * c[vv];
                cost[m * BN + n] = sqrtf(fmaxf(d2, 0.0f));
            }
        }
    }
    __syncthreads();

    // ---- Phase 3: Hungarian (Jonker-Volgenant, e-maxx variant) -------------
    const double INF = 1e300;
    for (int j = tid; j <= BN; j += BN) {   // tid 0 also handles j=128
        pA[j] = 0; wayA[j] = 0; u[j] = 0.0; v[j] = 0.0;
    }
    __syncthreads();

    for (int i = 1; i <= BN; ++i) {
        if (tid == 0) { pA[0] = i; ctrl[0] = 0; }
        for (int j = tid; j <= BN; j += BN) { minv[j] = INF; usedA[j] = 0; }
        __syncthreads();

        for (;;) {
            const int j0 = ctrl[0];
            if (tid == 0) usedA[j0] = 1;
            __syncthreads();
            const int i0 = pA[j0];

            // relax free columns (thread t owns column t+1)
            {
                const int j = tid + 1;
                if (!usedA[j]) {
                    double cur = (double)cost[(i0 - 1) * BN + (j - 1)] - u[i0] - v[j];
                    if (cur < minv[j]) { minv[j] = cur; wayA[j] = j0; }
                }
            }
            __syncthreads();

            // parallel argmin over free columns
            {
                const int j = tid + 1;
                redv[tid] = usedA[j] ? INF : minv[j];
                redj[tid] = j;
            }
            __syncthreads();
            for (int s = 64; s > 0; s >>= 1) {
                if (tid < s) {
                    if (redv[tid + s] < redv[tid]) {
                        redv[tid] = redv[tid + s];
                        redj[tid] = redj[tid + s];
                    }
                }
                __syncthreads();
            }
            const double delta = redv[0];
            const int    j1    = redj[0];

            // apply delta (used: u[p[j]]+=d, v[j]-=d ; free: minv[j]-=d)
            for (int j = tid; j <= BN; j += BN) {
                if (usedA[j]) { u[pA[j]] += delta; v[j] -= delta; }
                else          { minv[j] -= delta; }
            }
            if (tid == 0) ctrl[0] = j1;
            __syncthreads();

            if (pA[j1] == 0) break;   // uniform: all threads see same j1
        }

        // augment along alternating path (scalar)
        if (tid == 0) {
            int j0 = ctrl[0];
            while (j0) { int j1 = wayA[j0]; pA[j0] = pA[j1]; j0 = j1; }
        }
        __syncthreads();
    }

    // ---- Phase 4: loss terms (thread t = column n; row r = p[n+1]-1) -------
    {
        const int n = tid;
        const int r = pA[n + 1] - 1;

        const float* o_r = ob_base + (size_t)r * OC;   // output row  (rows idx)
        const float* o_n = ob_base + (size_t)n * OC;   // output row  (cols idx)
        const float* t_r = tb_base + (size_t)r * TC;   // target row  (rows idx)
        const float* t_n = tb_base + (size_t)n * TC;   // target row  (cols idx)

        // mse: output[rows] boxes vs target[cols] boxes, mean over 3 dims
        float d0 = o_r[1] - t_n[1];
        float d1 = o_r[2] - t_n[2];
        float d2 = o_r[3] - t_n[3];
        float mse = (d0 * d0 + d1 * d1 + d2 * d2) * (1.0f / 3.0f);

        // bce: p = output[cols,0], t = target[rows,0]  (swap mirrors reference)
        float pv  = o_n[0];
        float tv  = t_r[0];
        float lp  = fmaxf(logf(pv),     -100.0f);
        float l1p = fmaxf(log1pf(-pv),  -100.0f);
        float bce = -(tv * lp + (1.0f - tv) * l1p);

        // ce over 17 logits (reference slices [...,4:]): logits at rows, class at cols
        float mx = o_r[4];
        #pragma unroll
        for (int k = 1; k < NCLS; ++k) mx = fmaxf(mx, o_r[4 + k]);
        float se = 0.0f;
        #pragma unroll
        for (int k = 0; k < NCLS; ++k) se += expf(o_r[4 + k] - mx);
        float lse = logf(se) + mx;
        int   cls = (int)t_n[4];
        float ce  = lse - o_r[4 + cls];

        redv[tid] = (double)(ALPHA * bce + BETA * mse + GAMMA * ce);
    }
    __syncthreads();
    for (int s = 64; s > 0; s >>= 1) {
        if (tid < s) redv[tid] += redv[tid + s];
        __syncthreads();
    }
    if (tid == 0) batch_loss[b] = redv[0];
}

// Deterministic final sum (fixed order) and /B
__global__ void finalize_kernel(const double* __restrict__ batch_loss,
                                float* __restrict__ out)
{
    double s = 0.0;
    for (int b = 0; b < 64; ++b) s += batch_loss[b];
    out[0] = (float)(s / 64.0);
}

extern "C" void kernel_launch(void* const* d_in, const int* in_sizes, int n_in,
                              void* d_out, int out_size, void* d_ws, size_t ws_size,
                              hipStream_t stream)
{
    (void)in_sizes; (void)n_in; (void)out_size; (void)ws_size;
    const float* outp = (const float*)d_in[0];   // [64,128,21] f32
    const float* tgt  = (const float*)d_in[1];   // [64,128,6]  f32
    double* batch_loss = (double*)d_ws;          // 64 doubles of scratch

    hungarian_loss_kernel<<<64, 128, SMEM_BYTES, stream>>>(outp, tgt, batch_loss);
    finalize_kernel<<<1, 1, 0, stream>>>(batch_loss, (float*)d_out);
}